// StimulusLayer_12953621364760
// MI455X (gfx1250) — compile-verified
//
#include <hip/hip_runtime.h>
#include <math.h>

// Problem constants (from the reference):
constexpr int kB = 128;    // batch rows of x
constexpr int kI = 256;    // inner (feature) dim
constexpr int kO = 4096;   // stimulus rows
constexpr float kDecay = 0.95f;

// Tiling
constexpr int BM = 32;           // x rows per block
constexpr int BN = 64;           // stim rows per block
constexpr int TX = 16, TY = 16;  // 256 threads = 8 waves (wave32)
constexpr int RM = BM / TY;      // 2 rows per thread
constexpr int RN = BN / TX;      // 4 cols per thread
constexpr int LSTR = kI + 4;     // padded LDS row stride (floats); 260*4B keeps 16B align,
                                 // 260 % 64 == 4 -> conflict-free float4 reads

// ---- gfx1250 async copy: global -> LDS, 16 bytes per lane, tracked by ASYNCcnt ----
__device__ __forceinline__ void async_copy_b128(const float* gsrc, unsigned lds_byte_addr) {
  asm volatile("global_load_async_to_lds_b128 %0, %1, off"
               :: "v"(lds_byte_addr), "v"((unsigned long long)(const void*)gsrc)
               : "memory");
}

__device__ __forceinline__ void wait_async_zero() {
  asm volatile("s_wait_asynccnt 0" ::: "memory");
}

__global__ __launch_bounds__(TX * TY)
void stimulus_linf_kernel(const float* __restrict__ x,
                          const float* __restrict__ stim,
                          const float* __restrict__ a,
                          const float* __restrict__ bgain,
                          const float* __restrict__ state,
                          float* __restrict__ out) {
  __shared__ float lx[BM * LSTR];  // 32 x-rows,  padded
  __shared__ float ls[BN * LSTR];  // 64 s-rows,  padded

  const int tid = threadIdx.x;
  const int tx = tid & (TX - 1);
  const int ty = tid >> 4;
  const int m0 = blockIdx.y * BM;  // first x row of this block
  const int n0 = blockIdx.x * BN;  // first stim row of this block

  const unsigned lx_base = (unsigned)(unsigned long long)(const void*)lx;
  const unsigned ls_base = (unsigned)(unsigned long long)(const void*)ls;

  // ---------------- async stage: whole K for both tiles ----------------
  // x tile: BM rows * (kI/4)=64 float4 chunks = 2048 chunks, 8 per thread
  constexpr int CPR = kI / 4;  // 64 float4 chunks per row
  #pragma unroll
  for (int it = 0; it < (BM * CPR) / (TX * TY); ++it) {
    const int q   = tid + it * (TX * TY);
    const int row = q >> 6;              // q / CPR
    const int c4  = (q & (CPR - 1)) * 4; // float index within row
    async_copy_b128(x + (size_t)(m0 + row) * kI + c4,
                    lx_base + (unsigned)((row * LSTR + c4) * 4));
  }
  // stim tile: BN rows * 64 chunks = 4096 chunks, 16 per thread
  #pragma unroll
  for (int it = 0; it < (BN * CPR) / (TX * TY); ++it) {
    const int q   = tid + it * (TX * TY);
    const int row = q >> 6;
    const int c4  = (q & (CPR - 1)) * 4;
    async_copy_b128(stim + (size_t)(n0 + row) * kI + c4,
                    ls_base + (unsigned)((row * LSTR + c4) * 4));
  }

  wait_async_zero();   // each wave drains its own ASYNCcnt
  __syncthreads();     // then all waves' data is visible in LDS

  // ---------------- compute: L-inf distance, register-tiled ----------------
  float acc[RM][RN];
  #pragma unroll
  for (int r = 0; r < RM; ++r)
    #pragma unroll
    for (int c = 0; c < RN; ++c)
      acc[r][c] = 0.0f;  // |diff| >= 0

  #pragma unroll 4
  for (int k = 0; k < kI; k += 4) {
    float4 xv[RM], sv[RN];
    #pragma unroll
    for (int r = 0; r < RM; ++r)
      xv[r] = *(const float4*)&lx[(ty + r * TY) * LSTR + k];
    #pragma unroll
    for (int c = 0; c < RN; ++c)
      sv[c] = *(const float4*)&ls[(tx + c * TX) * LSTR + k];

    #pragma unroll
    for (int r = 0; r < RM; ++r) {
      #pragma unroll
      for (int c = 0; c < RN; ++c) {
        float m = acc[r][c];
        m = fmaxf(m, fabsf(xv[r].x - sv[c].x));
        m = fmaxf(m, fabsf(xv[r].y - sv[c].y));
        m = fmaxf(m, fabsf(xv[r].z - sv[c].z));
        m = fmaxf(m, fabsf(xv[r].w - sv[c].w));
        acc[r][c] = m;
      }
    }
  }

  // ---------------- epilogue: sigmoid((a - d) * b), decayed state ----------------
  #pragma unroll
  for (int c = 0; c < RN; ++c) {
    const int n = n0 + tx + c * TX;
    const float av = a[n];
    const float bv = bgain[n];
    #pragma unroll
    for (int r = 0; r < RM; ++r) {
      const int m = m0 + ty + r * TY;
      const float t = (av - acc[r][c]) * bv;
      const float sig = 1.0f / (1.0f + __expf(-t));
      const size_t idx = (size_t)m * kO + n;
      out[idx] = state[idx] * kDecay + sig;
    }
  }
}

extern "C" void kernel_launch(void* const* d_in, const int* in_sizes, int n_in,
                              void* d_out, int out_size, void* d_ws, size_t ws_size,
                              hipStream_t stream) {
  const float* x     = (const float*)d_in[0];
  const float* stim  = (const float*)d_in[1];
  const float* a     = (const float*)d_in[2];
  const float* bgain = (const float*)d_in[3];
  const float* state = (const float*)d_in[4];
  float* out = (float*)d_out;

  dim3 grid(kO / BN, kB / BM);   // 64 x 4 = 256 workgroups
  dim3 block(TX * TY);           // 256 threads = 8 wave32
  stimulus_linf_kernel<<<grid, block, 0, stream>>>(x, stim, a, bgain, state, out);
}